// ManualFeature_rot_3702261809447
// MI455X (gfx1250) — compile-verified
//
#include <hip/hip_runtime.h>
#include <math.h>

typedef float v2f __attribute__((ext_vector_type(2)));
typedef float v8f __attribute__((ext_vector_type(8)));

#define B_ 8
#define N_ 16384
#define L_ 605
#define GYZ 55          // GRID[1]*GRID[2] = 11*5
#define GZ 5            // GRID[2]
#define ANG 8
#define FEAT_STRIDE 16  // 15 counters padded to 16
#define OUTC 23         // 15 distance + 8 rotation channels
#define NT_V 38         // ceil(605/16) voxel tiles
#define NBIN 17         // distance bins 0..16 (16 = discard)

// ---------------- kernel 0: zero the u32 accumulators in workspace ----------
__global__ void k_zero(unsigned* __restrict__ ws, int nwords) {
    int i = blockIdx.x * blockDim.x + threadIdx.x;
    if (i < nwords) ws[i] = 0u;
}

// ---------------- kernel 1: distance feature via V_WMMA_F32_16X16X4_F32 -----
// grid = B_*NT_V workgroups, 256 threads (8 waves). Wave w handles point
// chunk [w*2048, (w+1)*2048) against this block's 16-voxel tile.
// Each lane bins its 8 D-elements per WMMA into a private 17-bin LDS
// histogram (one no-return ds_add per pair); cumulative counts are rebuilt
// with a single running sum at the end.
__global__ void k_feat(const float* __restrict__ pcd, unsigned* __restrict__ feat) {
    __shared__ unsigned lh[256 * NBIN];      // 17 KB, lane-private bins
    unsigned* lhist = lh + threadIdx.x * NBIN;
#pragma unroll
    for (int j = 0; j < NBIN; ++j) lhist[j] = 0u;   // private: no barrier needed

    const int b    = blockIdx.x / NT_V;
    const int vt   = blockIdx.x % NT_V;
    const int lane = threadIdx.x & 31;
    const int wave = threadIdx.x >> 5;
    const int col  = lane & 15;

    // Voxel (column) owned by this lane.
    const int v = vt * 16 + col;
    const bool vvalid = (v < L_);
    float lx, ly, lz;
    if (vvalid) {
        int ix = v / GYZ; int rem = v - ix * GYZ;
        int iy = rem / GZ; int iz = rem - iy * GZ;
        lx = -5.0f + (float)ix; ly = -5.0f + (float)iy; lz = -2.0f + (float)iz;
    } else {
        lx = 1.0e4f; ly = 1.0e4f; lz = 1.0e4f;   // pad voxel: lands in bin 16
    }
    const float ll2 = lx * lx + ly * ly + lz * lz;

    // B matrix (4x16): rows [lx, ly, lz, 1]. 32-bit layout:
    // V0: lanes0-15 = K0 row, lanes16-31 = K2 row; V1: K1 / K3 rows.
    v2f bm;
    bm.x = (lane < 16) ? lx : lz;
    bm.y = (lane < 16) ? ly : 1.0f;

    // C matrix: |loc|^2 broadcast down each column.
    v8f cm;
#pragma unroll
    for (int i = 0; i < 8; ++i) cm[i] = ll2;

    const float* pb = pcd + (size_t)b * (N_ * 3);
    const int base = wave * (N_ / 8);

    for (int t = 0; t < (N_ / 8) / 16; ++t) {
        const int n = base + t * 16 + col;
        const float* p = pb + n * 3;
        const float px = p[0], py = p[1], pz = p[2];
        const float pn2 = px * px + py * py + pz * pz;

        // A matrix (16x4): rows [-2px, -2py, -2pz, |p|^2]. 32-bit layout:
        // lanes0-15 hold {K0,K1}; lanes16-31 hold {K2,K3} for the same row m.
        v2f am;
        am.x = (lane < 16) ? (-2.0f * px) : (-2.0f * pz);
        am.y = (lane < 16) ? (-2.0f * py) : pn2;

        // d2[m=point, n=voxel] = |p|^2 + |loc|^2 - 2 p.loc  in one WMMA
        v8f d = __builtin_amdgcn_wmma_f32_16x16x4_f32(
            false, am, false, bm, (short)0, cm, false, false);

#pragma unroll
        for (int i = 0; i < 8; ++i) {
            // raw v_sqrt_f32 (1 instr) instead of the IEEE-precise expansion
            float s = __builtin_amdgcn_sqrtf(fmaxf(d[i], 0.0f));
            s = fminf(ceilf(s), 16.0f);
            atomicAdd(&lhist[(int)s], 1u);   // lane-private bin: ds_add_u32
        }
    }

    // Rebuild cumulative counts: cnt[k] = #(d <= k+1) = sum bins 0..k+1.
    int cnt[15];
    unsigned running = 0;
#pragma unroll
    for (int j = 0; j <= 15; ++j) {
        running += lhist[j];
        if (j >= 1) cnt[j - 1] = (int)running;
    }

    // Lanes L and L+16 hold the same voxel (point rows 0-7 / 8-15): combine.
#pragma unroll
    for (int k = 0; k < 15; ++k)
        cnt[k] += __shfl_xor(cnt[k], 16, 32);

    if (lane < 16 && vvalid) {
        unsigned* dst = feat + ((size_t)(b * L_ + v)) * FEAT_STRIDE;
#pragma unroll
        for (int k = 0; k < 15; ++k)
            if (cnt[k]) atomicAdd(&dst[k], (unsigned)cnt[k]);
    }
}

// ---------------- kernel 2: rotation histogram (LDS hist + global merge) ----
__global__ void k_rot(const float* __restrict__ pcd, unsigned* __restrict__ rotc) {
    // cos/sin of (pi/8*a - pi/2), a = 0..7 (float32)
    const float CT[8] = { -4.37113883e-08f, 0.38268343f, 0.70710678f, 0.92387953f,
                           1.0f,            0.92387953f, 0.70710678f, 0.38268343f };
    const float ST[8] = { -1.0f, -0.92387953f, -0.70710678f, -0.38268343f,
                           0.0f,  0.38268343f,  0.70710678f,  0.92387953f };

    __shared__ unsigned hist[ANG * L_];
    for (int i = threadIdx.x; i < ANG * L_; i += blockDim.x) hist[i] = 0u;
    __syncthreads();

    const int RCH = 8;
    const int b  = blockIdx.x / RCH;
    const int ch = blockIdx.x % RCH;
    const float* pb = pcd + (size_t)b * (N_ * 3);
    const int base = ch * (N_ / RCH);

    for (int j = 0; j < (N_ / RCH) / 256; ++j) {
        const int n = base + j * 256 + threadIdx.x;
        const float* p = pb + n * 3;
        const float px = p[0], py = p[1], pz = p[2];
        const float fz = floorf(pz + 2.5f);
#pragma unroll
        for (int a = 0; a < ANG; ++a) {
            const float fx = floorf(CT[a] * px - ST[a] * py + 5.5f);
            const float fy = floorf(ST[a] * px + CT[a] * py + 5.5f);
            const int idx = (int)(fz + fy * (float)GZ + fx * (float)GYZ);
            if (idx >= 0 && idx < L_) atomicAdd(&hist[a * L_ + idx], 1u);
        }
    }
    __syncthreads();

    for (int i = threadIdx.x; i < ANG * L_; i += blockDim.x) {
        const unsigned h = hist[i];
        if (h) atomicAdd(&rotc[(size_t)b * (ANG * L_) + i], h);
    }
}

// ---------------- kernel 3: scale + interleave into the output -------------
__global__ void k_final(const unsigned* __restrict__ feat,
                        const unsigned* __restrict__ rotc,
                        float* __restrict__ out) {
    const int t = blockIdx.x * blockDim.x + threadIdx.x;
    if (t >= B_ * L_) return;
    const int b = t / L_, l = t - b * L_;
    const float inv = 1.0f / (float)N_;
    float* o = out + (size_t)t * OUTC;
    const unsigned* f = feat + (size_t)t * FEAT_STRIDE;
#pragma unroll
    for (int k = 0; k < 15; ++k) o[k] = (float)f[k] * inv;
#pragma unroll
    for (int a = 0; a < ANG; ++a)
        o[15 + a] = (float)rotc[((size_t)b * ANG + a) * L_ + l] * inv;
}

extern "C" void kernel_launch(void* const* d_in, const int* in_sizes, int n_in,
                              void* d_out, int out_size, void* d_ws, size_t ws_size,
                              hipStream_t stream) {
    const float* pcd = (const float*)d_in[0];
    float* out = (float*)d_out;

    unsigned* feat = (unsigned*)d_ws;                       // B*L*16 u32
    unsigned* rotc = feat + (size_t)B_ * L_ * FEAT_STRIDE;  // B*ANG*L u32
    const int nwords = B_ * L_ * FEAT_STRIDE + B_ * ANG * L_;

    k_zero <<<(nwords + 255) / 256, 256, 0, stream>>>((unsigned*)d_ws, nwords);
    k_feat <<<B_ * NT_V, 256, 0, stream>>>(pcd, feat);
    k_rot  <<<B_ * 8,    256, 0, stream>>>(pcd, rotc);
    k_final<<<(B_ * L_ + 255) / 256, 256, 0, stream>>>(feat, rotc, out);
}